// CustomGenGaps_71536975283066
// MI455X (gfx1250) — compile-verified
//
#include <hip/hip_runtime.h>
#include <stdint.h>

#define H_DIM    5
#define THETA_F  0.0001f
#define RHO_F    1e-05f
#define NTHREADS 256
#define RECCAP   4096   // cap on outer-step records (expected ~1.6k before cutoff)

// ---------------- threefry-2x32 (JAX PRNG) ----------------
__device__ __forceinline__ uint32_t rotl32(uint32_t v, int n) {
  return (v << n) | (v >> (32 - n));
}

__device__ __forceinline__ void tf2x32(uint32_t k0, uint32_t k1,
                                       uint32_t c0, uint32_t c1,
                                       uint32_t& o0, uint32_t& o1) {
  uint32_t ks2 = k0 ^ k1 ^ 0x1BD11BDAu;
  uint32_t x0 = c0 + k0;
  uint32_t x1 = c1 + k1;
#define TF_GROUP(ra, rb, rc, rd, ia, ib)                 \
  x0 += x1; x1 = rotl32(x1, ra); x1 ^= x0;               \
  x0 += x1; x1 = rotl32(x1, rb); x1 ^= x0;               \
  x0 += x1; x1 = rotl32(x1, rc); x1 ^= x0;               \
  x0 += x1; x1 = rotl32(x1, rd); x1 ^= x0;               \
  x0 += (ia); x1 += (ib);
  TF_GROUP(13, 15, 26, 6,  k1,  ks2 + 1u)
  TF_GROUP(17, 29, 16, 24, ks2, k0 + 2u)
  TF_GROUP(13, 15, 26, 6,  k0,  k1 + 3u)
  TF_GROUP(17, 29, 16, 24, k1,  ks2 + 4u)
  TF_GROUP(13, 15, 26, 6,  ks2, k0 + 5u)
#undef TF_GROUP
  o0 = x0; o1 = x1;
}

__device__ __forceinline__ float bits_to_unit(uint32_t bits) {
  // JAX float32 uniform in [0,1): (bits>>9)|0x3f800000 bitcast, minus 1
  return __uint_as_float((bits >> 9) | 0x3f800000u) - 1.0f;
}

// subkey for scan step s:  keys = jax.random.split(kA, MS)
__device__ __forceinline__ void subkey_of(int s, int MS,
                                          uint32_t kA0, uint32_t kA1,
                                          uint32_t& sa, uint32_t& sb) {
  const int i2 = 2 * s;
  uint32_t d0, d1;
  if (i2 < MS) {  // first half of split() outputs
    tf2x32(kA0, kA1, (uint32_t)i2,       (uint32_t)(MS + i2),     sa, d0);
    tf2x32(kA0, kA1, (uint32_t)(i2 + 1), (uint32_t)(MS + i2 + 1), sb, d1);
  } else {        // second half
    tf2x32(kA0, kA1, (uint32_t)(i2 - MS),     (uint32_t)i2,       d0, sa);
    tf2x32(kA0, kA1, (uint32_t)(i2 + 1 - MS), (uint32_t)(i2 + 1), d1, sb);
  }
}

// ---------------- kernel ----------------
__global__ __launch_bounds__(NTHREADS)
void gen_gaps_gfx1250(const float* __restrict__ wptr,
                      const float* __restrict__ bptr,
                      const float* __restrict__ vptr,
                      float* __restrict__ out, int k) {
  extern __shared__ float smem[];
  __shared__ int   s_cnt[NTHREADS];
  __shared__ int   s_base[NTHREADS];
  __shared__ float s_part[NTHREADS];
  __shared__ float s_pbase[NTHREADS];
  __shared__ int   s_Tstar, s_Rstar, s_totalinner;

  const int tid = threadIdx.x;
  const int K2  = k + 2;
  const int MS  = 2 * K2 + 256;            // max_steps
  const int pad = (K2 + 3) & ~3;
  const int MSp = (MS + 3) & ~3;

  float*  ta  = smem;                      // K2 floats (histogram, zeroed)
  float*  ecs = smem + pad;                // MS floats: |ecs|=e/c, sign = inner flag
  float4* rec = (float4*)(smem + pad + MSp); // RECCAP+1 outer records {u1, z, rcp, -}

  // ---- scalar parameters ----
  float wf[H_DIM], bf[H_DIM], vf[H_DIM];
#pragma unroll
  for (int i = 0; i < H_DIM; ++i) {
    wf[i] = wptr[i];
    vf[i] = vptr[i];
    bf[i] = (i == 0) ? 0.0f : bptr[i - 1];
  }
  float v0 = 0.0f;
#pragma unroll
  for (int i = 0; i < H_DIM; ++i) v0 -= bf[i] * vf[i];
  const float cc     = THETA_F + RHO_F;
  const float inv_cc = 1.0f / cc;
  const float pp     = THETA_F / (THETA_F + RHO_F);   // ~0.909 -> ~91% inner
  const float vb0    = vf[0] * bf[0];
  const float vw0    = vf[0] * wf[0];
  const float t0c    = -bf[0] / wf[0];
  float rxv = v0;
#pragma unroll
  for (int i = 0; i < H_DIM; ++i) rxv += fmaxf(wf[i] * t0c + bf[i], 0.0f) * vf[i];
  const float gth = fmaxf(0.0f, rxv);      // gate threshold (searchsorted >= 2)

  // ---- JAX key derivation: key(1234) -> split -> (kA, kB) ----
  uint32_t p0, p1, q0, q1;
  tf2x32(0u, 1234u, 0u, 2u, p0, p1);
  tf2x32(0u, 1234u, 1u, 3u, q0, q1);
  const uint32_t kA0 = p0, kA1 = q0;       // scan keys
  const uint32_t kB0 = p1, kB1 = q1;       // z0 key
  uint32_t z0b, dz;
  tf2x32(kB0, kB1, 0u, 0u, z0b, dz);
  const float z0 = -logf(bits_to_unit(z0b));

  // thread's contiguous step range
  const int L  = (MS + NTHREADS - 1) / NTHREADS;
  const int lo = tid * L;
  const int hi = (lo + L < MS) ? (lo + L) : MS;

  // ==== Walk A (parallel): ec = (-log u3)/c with inner flag in sign bit ====
  for (int i = tid; i < K2; i += NTHREADS) ta[i] = 0.0f;   // zero histogram
  int cnt = 0;
  if (lo < hi) {
    float u4prev = 0.0f;
    if (lo > 0) {                           // predecessor's u4 (for inner_lo)
      uint32_t sa, sb, ub2, ub4;
      subkey_of(lo - 1, MS, kA0, kA1, sa, sb);
      tf2x32(sa, sb, 1u, 3u, ub2, ub4);
      u4prev = bits_to_unit(ub4);
    }
    for (int t = lo; t < hi; ++t) {
      uint32_t sa, sb, u1b, u3b, u2b, u4b;
      subkey_of(t, MS, kA0, kA1, sa, sb);
      tf2x32(sa, sb, 0u, 2u, u1b, u3b);     // uniform(subkey,(4,)) layout
      tf2x32(sa, sb, 1u, 3u, u2b, u4b);
      const float ec    = -logf(bits_to_unit(u3b)) * inv_cc;  // e/c > 0 strictly
      const bool  inner = (t > 0) && (u4prev < pp);
      ecs[t] = inner ? -ec : ec;
      cnt += inner ? 1 : 0;
      u4prev = bits_to_unit(u4b);
    }
  }
  s_cnt[tid] = cnt;
  __syncthreads();

  // ==== exclusive scan of inner counts (thread 0, 256 partials) ====
  if (tid == 0) {
    int run = 0;
    for (int i = 0; i < NTHREADS; ++i) { s_base[i] = run; run += s_cnt[i]; }
    s_totalinner = run;
    s_Tstar = MS - 1;                       // sentinel: process everything
    s_Rstar = -1;
  }
  __syncthreads();

  // ==== Walk B (parallel): compact outer records, find cutoff T*/R* ====
  {
    int j = s_base[tid];
    for (int t = lo; t < hi; ++t) {
      const bool inner = (ecs[t] < 0.0f);
      if (inner) {
        if (j == K2 - 1) { s_Tstar = t; s_Rstar = t + 1 - K2; }  // unique thread
        ++j;
      } else {
        const int rank = t - j;             // #outer before t == t - j_t
        if (rank < RECCAP) {
          uint32_t sa, sb, u1b, u3b, u2b, u4b;
          subkey_of(t, MS, kA0, kA1, sa, sb);
          tf2x32(sa, sb, 0u, 2u, u1b, u3b);
          tf2x32(sa, sb, 1u, 3u, u2b, u4b);
          rec[rank].x = bits_to_unit(u1b);            // u1
          rec[rank].y = -logf(bits_to_unit(u2b));     // z = -log u2
        }
      }
    }
  }
  __syncthreads();

  int Tlim = s_Tstar;
  int R    = s_Rstar;
  if (R < 0) R = MS - s_totalinner;         // cutoff never reached
  if (R > RECCAP) R = RECCAP;

  // ==== Phase B (lane 0): the ONLY serial chain -- x over outer steps ====
  if (tid == 0) {
    float x = (z0 >= gth) ? ((z0 - v0 - vb0) / vw0) : 0.0f;   // inverse(z0)
    float4 rc = rec[0];
    for (int r = 0; r < R; ++r) {
      const float4 nx = rec[r + 1];         // padded slot: unconditional prefetch
      const float u_x = rc.x * x;
      float racc = v0;
#pragma unroll
      for (int i = 0; i < H_DIM; ++i)
        racc += fmaxf(wf[i] * u_x + bf[i], 0.0f) * vf[i];
      const float xv = rc.y + racc;
      float rcp;
      if (xv >= gth) {
        const float num = xv - v0 - vb0;
        x   = num / vw0;                    // exact divide, on-chain (short)
        rcp = vw0 / num;                    // divide hangs OFF the x-chain
      } else {
        x   = 0.0f;
        rcp = __builtin_inff();             // matches e/(0*c) -> +inf
      }
      ((float*)&rec[r])[2] = rcp;           // rec[r].z
      rc = nx;
    }
  }
  __syncthreads();

  // ==== Phase C1 (parallel): per-range partial sums of d_t = ec * rcp[seg] ====
  {
    const int hi2 = (hi < Tlim + 1) ? hi : (Tlim + 1);
    int j = s_base[tid];
    float sum = 0.0f;
    for (int t = lo; t < hi2; ++t) {
      const float es = ecs[t];
      const bool inner = (es < 0.0f);
      int seg = t - j - (inner ? 1 : 0);    // governing outer rank
      if (seg >= RECCAP) seg = RECCAP - 1;
      if (inner) ++j;
      sum += fabsf(es) * ((const float*)&rec[seg])[2];
    }
    s_part[tid] = sum;
  }
  __syncthreads();
  if (tid == 0) {
    float run = 0.0f;
    for (int i = 0; i < NTHREADS; ++i) { s_pbase[i] = run; run += s_part[i]; }
  }
  __syncthreads();

  // ==== Phase C2 (parallel): scatter ta[j] = pos at inner steps ====
  {
    const int hi2 = (hi < Tlim + 1) ? hi : (Tlim + 1);
    int j = s_base[tid];
    float pos = s_pbase[tid];
    for (int t = lo; t < hi2; ++t) {
      const float es = ecs[t];
      const bool inner = (es < 0.0f);
      int seg = t - j - (inner ? 1 : 0);
      if (seg >= RECCAP) seg = RECCAP - 1;
      if (inner) { ta[j] = pos; ++j; }      // pre-update pos, as in reference
      pos += fabsf(es) * ((const float*)&rec[seg])[2];
    }
  }
  __syncthreads();

  // ==== Epilogue: diff in LDS, drain via CDNA5 async LDS->global stores ====
  float* diff_sh = (float*)rec;             // 4096-float staging (records dead)
  const int DC  = 4096;
  const int PER = DC / NTHREADS;            // 16 floats per thread
  const int ND  = (k + DC - 1) / DC;
  for (int d = 0; d < ND; ++d) {
    const int gb = d * DC;
    const int lb = tid * PER;
#pragma unroll
    for (int q = 0; q < PER; ++q) {
      const int i = gb + lb + q;
      diff_sh[lb + q] = (i < k) ? (ta[i + 2] - ta[i + 1]) : 0.0f;
    }
    __syncthreads();
#pragma unroll
    for (int q = 0; q < PER; q += 4) {
      const int i = gb + lb + q;
      if (i < k) {
        unsigned long long ga = (unsigned long long)(uintptr_t)(out + i);
        unsigned lds_off = (unsigned)(uintptr_t)(&diff_sh[lb + q]);
        asm volatile("global_store_async_from_lds_b128 %0, %1, off"
                     :: "v"(ga), "v"(lds_off) : "memory");
      }
    }
#if __has_builtin(__builtin_amdgcn_s_wait_asynccnt)
    __builtin_amdgcn_s_wait_asynccnt(0);
#else
    asm volatile("s_wait_asynccnt 0" ::: "memory");
#endif
    __syncthreads();                        // staging safe to reuse
  }
}

extern "C" void kernel_launch(void* const* d_in, const int* in_sizes, int n_in,
                              void* d_out, int out_size, void* d_ws, size_t ws_size,
                              hipStream_t stream) {
  const float* w = (const float*)d_in[0];   // (1,H) -> 5 floats
  const float* b = (const float*)d_in[1];   // (H-1,) -> 4 floats
  const float* v = (const float*)d_in[2];   // (H,1) -> 5 floats
  float* out = (float*)d_out;
  const int k   = out_size;                 // k == 16384
  const int K2  = k + 2;
  const int MS  = 2 * K2 + 256;
  const int pad = (K2 + 3) & ~3;
  const int MSp = (MS + 3) & ~3;
  const size_t shmem =
      (size_t)(pad + MSp + (RECCAP + 1) * 4) * sizeof(float); // ~257 KB < 320 KB WGP LDS
  gen_gaps_gfx1250<<<1, NTHREADS, shmem, stream>>>(w, b, v, out, k);
}